// DecoderCell_12790412608074
// MI455X (gfx1250) — compile-verified
//
#include <hip/hip_runtime.h>
#include <hip/hip_bf16.h>
#include <stdint.h>

// Problem constants (fixed by the reference)
#define BB 256
#define NN 2000
#define DD 128
#define HH 8
#define NT 125            // 16-node tiles per batch
#define TPB 4             // tiles per block (4 waves of 32)
#define CHUNK 125         // nodes per y-chunk
#define NCHUNK 16
#define XH 132            // LDS row stride in floats (128 + pad4): bank shift 4/row

typedef float v2f __attribute__((ext_vector_type(2)));
typedef float v4f __attribute__((ext_vector_type(4)));
typedef float v8f __attribute__((ext_vector_type(8)));

// workspace layout (in floats)
#define OFF_U      0                              // B*128*16  (U, zero-padded cols 8..15)
#define OFF_SCORES (OFF_U + BB*DD*16)             // B*N*8
#define OFF_STATS  (OFF_SCORES + BB*NN*HH)        // B*16 (max[8], invsum[8])
#define OFF_YPART  (OFF_STATS + BB*16)            // B*16*128*8
#define OFF_W2     (OFF_YPART + BB*NCHUNK*DD*HH)  // B*128

__device__ __forceinline__ void wait_asynccnt0() {
#if __has_builtin(__builtin_amdgcn_s_wait_asynccnt)
  __builtin_amdgcn_s_wait_asynccnt(0);
#else
  asm volatile("s_wait_asynccnt 0x0" ::: "memory");
#endif
}

// ---------------------------------------------------------------------------
// K1: per batch — q = ge@Wq_fixed + sc@Wq_step, then U[d][h] = sum_{j in head h} Wk1[d][j]*q[j]
// ---------------------------------------------------------------------------
__global__ __launch_bounds__(128) void k1_prep(
    const float* __restrict__ ge, const float* __restrict__ sc,
    const float* __restrict__ Wqf, const float* __restrict__ Wqs,
    const float* __restrict__ Wk1, float* __restrict__ ws)
{
  const int b = blockIdx.x;
  const int j = threadIdx.x;
  __shared__ float qlds[DD];

  const float* geb = ge + (size_t)b * DD;
  const float* scb = sc + (size_t)b * (DD + 2);
  float q = 0.f;
  #pragma unroll 4
  for (int d = 0; d < DD; ++d)      q += geb[d] * Wqf[d * DD + j];   // coalesced over j
  #pragma unroll 2
  for (int d = 0; d < DD + 2; ++d)  q += scb[d] * Wqs[d * DD + j];
  qlds[j] = q;
  __syncthreads();

  const int d = threadIdx.x;
  float acc[HH];
  #pragma unroll
  for (int h = 0; h < HH; ++h) acc[h] = 0.f;
  const float* row = Wk1 + (size_t)d * DD;
  #pragma unroll 4
  for (int jj = 0; jj < DD; ++jj) acc[jj >> 4] += row[jj] * qlds[jj];

  float* U = ws + OFF_U + ((size_t)b * DD + d) * 16;
  #pragma unroll
  for (int h = 0; h < HH; ++h) { U[h] = acc[h]; U[HH + h] = 0.f; }
}

// ---------------------------------------------------------------------------
// K2: scores[b][n][h] = 0.25 * (x_n . U_h) via V_WMMA_F32_16X16X4_F32
//     one wave per 16-node tile; x tile staged with async global->LDS (NT),
//     masked entries stored as -inf
// ---------------------------------------------------------------------------
__global__ __launch_bounds__(128) void k2_scores(
    const float* __restrict__ x, const unsigned char* __restrict__ mask,
    float* __restrict__ ws)
{
  __shared__ float Ulds[DD * 16];       // 8 KB, shared by block
  __shared__ float Xlds[TPB][16 * XH];  // 4 waves * 8.25 KB = 33 KB

  const int b    = blockIdx.y;
  const int tid  = threadIdx.x;
  const int wave = tid >> 5;
  const int lane = tid & 31;

  const float* U = ws + OFF_U + (size_t)b * DD * 16;
  for (int i = tid; i < DD * 16; i += 128) Ulds[i] = U[i];
  __syncthreads();

  const int tile = blockIdx.x * TPB + wave;
  if (tile >= NT) return;
  const int node0 = tile * 16;
  const float* xt = x + ((size_t)b * NN + node0) * DD;

  // stage 16 rows x 128 floats directly into LDS via async engine
  // (512 float4 total; per iteration 32 lanes x 16B = one coalesced 512B burst)
  #pragma unroll
  for (int i = 0; i < 16; ++i) {
    const int idx = i * 32 + lane;
    const int row = idx >> 5;
    const int cc  = idx & 31;
    uint64_t ga = (uint64_t)(uintptr_t)(xt + (size_t)row * DD + cc * 4);
    uint32_t la = (uint32_t)(uintptr_t)&Xlds[wave][row * XH + cc * 4];
    asm volatile("global_load_async_to_lds_b128 %0, %1, off th:TH_LOAD_NT"
                 :: "v"(la), "v"(ga) : "memory");
  }
  wait_asynccnt0();
  __builtin_amdgcn_wave_barrier();

  const int nn   = lane & 15;          // A: row m / B: col n
  const int koff = (lane >> 4) << 1;   // 0 (lanes 0-15) or 2 (lanes 16-31)
  const float* Xrow = &Xlds[wave][nn * XH];

  v8f c = {0.f, 0.f, 0.f, 0.f, 0.f, 0.f, 0.f, 0.f};
  #pragma unroll
  for (int t = 0; t < 32; ++t) {
    const int k = 4 * t + koff;
    v2f a;  a.x  = Xrow[k];                 a.y  = Xrow[k + 1];
    v2f bv; bv.x = Ulds[k * 16 + nn];       bv.y = Ulds[(k + 1) * 16 + nn];
    // D = A(16x4) * B(4x16) + C, fp32
    c = __builtin_amdgcn_wmma_f32_16x16x4_f32(
            false, a, false, bv, (short)0, c, false, false);
  }

  // C layout: lane L -> col n = L%16, rows m = 8*(L/16)+v
  if (nn < HH) {
    const int mbase = 8 * (lane >> 4);
    #pragma unroll
    for (int v = 0; v < 8; ++v) {
      const int node = node0 + mbase + v;
      float s = c[v] * 0.25f;                       // 1/sqrt(dh), dh=16
      if (mask[(size_t)b * NN + node]) s = -__builtin_inff();
      ws[OFF_SCORES + ((size_t)b * NN + node) * HH + nn] = s;
    }
  }
}

// ---------------------------------------------------------------------------
// K3: per (b,h) masked-softmax stats: max and 1/sum(exp)
// ---------------------------------------------------------------------------
__global__ __launch_bounds__(256) void k3_stats(float* __restrict__ ws)
{
  __shared__ float red[256 * HH];      // 8 KB
  const int b = blockIdx.x, t = threadIdx.x;
  const float* S = ws + OFF_SCORES + (size_t)b * NN * HH;

  float lmax[HH];
  #pragma unroll
  for (int h = 0; h < HH; ++h) lmax[h] = -__builtin_inff();
  for (int n = t; n < NN; n += 256) {
    #pragma unroll
    for (int h = 0; h < HH; ++h) lmax[h] = fmaxf(lmax[h], S[(size_t)n * HH + h]);
  }
  #pragma unroll
  for (int h = 0; h < HH; ++h) red[t * HH + h] = lmax[h];
  __syncthreads();
  for (int s2 = 128; s2 > 0; s2 >>= 1) {
    if (t < s2) {
      #pragma unroll
      for (int h = 0; h < HH; ++h)
        red[t * HH + h] = fmaxf(red[t * HH + h], red[(t + s2) * HH + h]);
    }
    __syncthreads();
  }
  float gmax[HH];
  #pragma unroll
  for (int h = 0; h < HH; ++h) gmax[h] = red[h];
  __syncthreads();

  float lsum[HH];
  #pragma unroll
  for (int h = 0; h < HH; ++h) lsum[h] = 0.f;
  for (int n = t; n < NN; n += 256) {
    #pragma unroll
    for (int h = 0; h < HH; ++h) lsum[h] += expf(S[(size_t)n * HH + h] - gmax[h]);
  }
  #pragma unroll
  for (int h = 0; h < HH; ++h) red[t * HH + h] = lsum[h];
  __syncthreads();
  for (int s2 = 128; s2 > 0; s2 >>= 1) {
    if (t < s2) {
      #pragma unroll
      for (int h = 0; h < HH; ++h) red[t * HH + h] += red[(t + s2) * HH + h];
    }
    __syncthreads();
  }
  if (t == 0) {
    #pragma unroll
    for (int h = 0; h < HH; ++h) {
      ws[OFF_STATS + b * 16 + h]      = gmax[h];
      ws[OFF_STATS + b * 16 + 8 + h]  = 1.f / red[h];
    }
  }
}

// ---------------------------------------------------------------------------
// K4: y partials — yp[b][chunk][d][h] = sum_{n in chunk} attn[n][h] * x[n][d]
// ---------------------------------------------------------------------------
__global__ __launch_bounds__(128) void k4_ypart(
    const float* __restrict__ x, float* __restrict__ ws)
{
  __shared__ float p[CHUNK * HH];      // 1000 floats
  const int b = blockIdx.y, chunk = blockIdx.x, t = threadIdx.x;

  const float* S  = ws + OFF_SCORES + ((size_t)b * NN + chunk * CHUNK) * HH;
  const float* st = ws + OFF_STATS + b * 16;
  for (int i = t; i < CHUNK * HH; i += 128) {
    int h = i & 7;
    p[i] = expf(S[i] - st[h]) * st[8 + h];   // exp(-inf - m) = 0 handles the mask
  }
  __syncthreads();

  float acc[HH];
  #pragma unroll
  for (int h = 0; h < HH; ++h) acc[h] = 0.f;
  const float* xb = x + ((size_t)b * NN + chunk * CHUNK) * DD + t;
  for (int n = 0; n < CHUNK; ++n) {
    float xv = __builtin_nontemporal_load(xb + (size_t)n * DD); // coalesced, NT stream
    #pragma unroll
    for (int h = 0; h < HH; ++h) acc[h] += p[n * HH + h] * xv;
  }
  float* yp = ws + OFF_YPART + (((size_t)b * NCHUNK + chunk) * DD + t) * HH;
  #pragma unroll
  for (int h = 0; h < HH; ++h) yp[h] = acc[h];
}

// ---------------------------------------------------------------------------
// K5: reduce y, then ctx = y@Wv (head-sliced), Q2 = ctx@Wout, w2 = Wk2@Q2
// ---------------------------------------------------------------------------
__global__ __launch_bounds__(128) void k5_small(
    const float* __restrict__ Wv, const float* __restrict__ Wout,
    const float* __restrict__ Wk2, float* __restrict__ ws)
{
  __shared__ float ylds[DD * HH];
  __shared__ float ctx[DD];
  __shared__ float q2[DD];
  const int b = blockIdx.x, t = threadIdx.x;

  float y[HH];
  #pragma unroll
  for (int h = 0; h < HH; ++h) y[h] = 0.f;
  const float* yp = ws + OFF_YPART + (size_t)b * NCHUNK * DD * HH;
  for (int cch = 0; cch < NCHUNK; ++cch) {
    #pragma unroll
    for (int h = 0; h < HH; ++h) y[h] += yp[((size_t)cch * DD + t) * HH + h];
  }
  #pragma unroll
  for (int h = 0; h < HH; ++h) ylds[t * HH + h] = y[h];
  __syncthreads();

  const int j = t, h = j >> 4;
  float cj = 0.f;
  for (int d = 0; d < DD; ++d) cj += ylds[d * HH + h] * Wv[(size_t)d * DD + j];
  ctx[j] = cj;
  __syncthreads();

  float qo = 0.f;
  for (int jj = 0; jj < DD; ++jj) qo += ctx[jj] * Wout[(size_t)jj * DD + t];
  q2[t] = qo;
  __syncthreads();

  float w = 0.f;
  const float* row = Wk2 + (size_t)t * DD;
  for (int o = 0; o < DD; ++o) w += row[o] * q2[o];
  ws[OFF_W2 + (size_t)b * DD + t] = w;
}

// ---------------------------------------------------------------------------
// K6: logits[b][n] = 10*tanh((x_n . w2)/sqrt(128)); masked -> -inf
//     one wave per node, coalesced NT float4 row loads + shuffle reduce
// ---------------------------------------------------------------------------
__global__ __launch_bounds__(256) void k6_logits(
    const float* __restrict__ x, const unsigned char* __restrict__ mask,
    const float* __restrict__ ws, float* __restrict__ out)
{
  const int b = blockIdx.y, chunk = blockIdx.x;
  const int wave = threadIdx.x >> 5, lane = threadIdx.x & 31;

  const float* w2 = ws + OFF_W2 + (size_t)b * DD;
  v4f wv = *((const v4f*)w2 + lane);        // hot, keep RT

  const int base = chunk * 250;
  for (int n = base + wave; n < base + 250; n += 8) {
    const float* xr = x + ((size_t)b * NN + n) * DD;
    v4f xv = __builtin_nontemporal_load((const v4f*)xr + lane);
    float dot = xv.x * wv.x + xv.y * wv.y + xv.z * wv.z + xv.w * wv.w;
    #pragma unroll
    for (int s = 16; s > 0; s >>= 1) dot += __shfl_xor(dot, s, 32);
    if (lane == 0) {
      float v = 10.0f * tanhf(dot * 0.08838834764831845f);  // CLIP * tanh(/sqrt(128))
      if (mask[(size_t)b * NN + n]) v = -__builtin_inff();
      out[(size_t)b * NN + n] = v;
    }
  }
}

// ---------------------------------------------------------------------------
extern "C" void kernel_launch(void* const* d_in, const int* in_sizes, int n_in,
                              void* d_out, int out_size, void* d_ws, size_t ws_size,
                              hipStream_t stream) {
  const float* x    = (const float*)d_in[0];          // node_embeddings [B,N,D]
  const float* ge   = (const float*)d_in[1];          // graph_embedding [B,D]
  const float* sc   = (const float*)d_in[2];          // step_context [B,1,D+2]
  const unsigned char* mask = (const unsigned char*)d_in[3]; // bool [B,N,1]
  const float* Wk1  = (const float*)d_in[4];
  const float* Wv   = (const float*)d_in[5];
  const float* Wk2  = (const float*)d_in[6];
  const float* Wqf  = (const float*)d_in[7];
  const float* Wout = (const float*)d_in[8];
  const float* Wqs  = (const float*)d_in[9];
  float* out = (float*)d_out;
  float* ws  = (float*)d_ws;

  k1_prep  <<<BB, 128, 0, stream>>>(ge, sc, Wqf, Wqs, Wk1, ws);
  k2_scores<<<dim3(32, BB), 128, 0, stream>>>(x, mask, ws);
  k3_stats <<<BB, 256, 0, stream>>>(ws);
  k4_ypart <<<dim3(NCHUNK, BB), 128, 0, stream>>>(x, ws);
  k5_small <<<BB, 128, 0, stream>>>(Wv, Wout, Wk2, ws);
  k6_logits<<<dim3(8, BB), 256, 0, stream>>>(x, mask, ws, out);
}